// DTCWTInverse_69707319214737
// MI455X (gfx1250) — compile-verified
//
#include <hip/hip_runtime.h>

// ---------------------------------------------------------------------------
// Inverse DTCWT (3 levels) for MI455X / gfx1250, all-f32, WMMA-based FIR.
//
// Every separable filter pass is a banded-matrix GEMM on
// V_WMMA_F32_16X16X4_F32:   D(16x16) = F(16xK) * X(Kx16)   (column pass)
//                           D(16x16) = X(16xK) * G(Kx16)   (row pass)
// K = 16 for the q-shift 2x-upsampling synthesis (reflection window fits
// exactly), K = 24 for the 7/5-tap level-1 filters.
//
// Banded filter matrices are built once per launch (tiny builder kernels,
// 192 KB L2-resident region). Hot kernels are templated on the fixed tensor
// dimensions so all strides are compile-time constants (IOFFSET folding,
// SADDR + 32-bit voffset addressing); c2q is fused into the B-fragment
// loads using the even/odd row-pair structure (w0 is always even).
// ---------------------------------------------------------------------------

typedef __attribute__((ext_vector_type(2))) float v2f;
typedef __attribute__((ext_vector_type(8))) float v8f;

#define SC_HALF_SQRT2 0.70710678118654752f

__device__ __forceinline__ int refl(int i, int n) {
  i = (i < 0) ? (-1 - i) : i;
  i = (i >= n) ? (2 * n - 1 - i) : i;
  return i;
}

__device__ __forceinline__ int iclamp(int v, int lo, int hi) {
  return v < lo ? lo : (v > hi ? hi : v);
}

__device__ __forceinline__ v8f wmma4(v2f a, v2f b, v8f c) {
  // D = A(16x4,f32) * B(4x16,f32) + C(16x16,f32)
  return __builtin_amdgcn_wmma_f32_16x16x4_f32(false, a, false, b, (short)0, c,
                                               false, false);
}

// Synthesis coefficient for colifilt/rowifilt (10-tap q-shift, 2x upsample).
__device__ __forceinline__ float ifilt_coef(int o, int kg, int n,
                                            const float (&ha)[10],
                                            const float (&hb)[10], bool hp) {
  const int i = o >> 2, s = o & 3;
  const int base = 4 + ((((s & 1) != 0) != hp) ? 1 : 0);
  float acc = 0.f;
#pragma unroll
  for (int m = 0; m < 5; ++m) {
    const int L = refl(2 * i + base - 2 * m, n);
    const float c = (s == 0)   ? ha[2 * m]
                    : (s == 1) ? hb[2 * m]
                    : (s == 2) ? ha[2 * m + 1]
                               : hb[2 * m + 1];
    acc += (L == kg) ? c : 0.f;
  }
  return acc;
}

// Coefficient for plain symmetric-extension FIR (colfilter/rowfilter).
template <int NT>
__device__ __forceinline__ float filt_coef(int o, int kg, int n,
                                           const float (&h)[NT]) {
  float acc = 0.f;
#pragma unroll
  for (int m = 0; m < NT; ++m) {
    const int L = refl(o + (NT >> 1) - m, n);
    acc += (L == kg) ? h[m] : 0.f;
  }
  return acc;
}

// ===========================================================================
// Builder kernels (cold path): banded filter matrices, shared by all maps.
// ===========================================================================

// Column-pass q-shift matrix: M[ty][f][row][k], 16x16, f=0 low / f=1 high.
__global__ __launch_bounds__(256) void build_ifilt_colF(
    const float* __restrict__ g0a, const float* __restrict__ g0b,
    const float* __restrict__ g1a, const float* __restrict__ g1b,
    float* __restrict__ M, int r) {
  const int ty = blockIdx.x, f = blockIdx.y;
  float ha[10], hb[10];
#pragma unroll
  for (int m = 0; m < 10; ++m) {
    ha[m] = f ? g1b[m] : g0b[m];
    hb[m] = f ? g1a[m] : g0a[m];
  }
  const int o0 = ty << 4;
  const int w0 = iclamp((o0 >> 1) - 4, 0, r - 16);
  const int idx = threadIdx.x;
  const int row = idx >> 4, k = idx & 15;
  M[(((size_t)ty * 2 + f) * 16 + row) * 16 + k] =
      ifilt_coef(o0 + row, w0 + k, r, ha, hb, f != 0);
}

// Row-pass q-shift matrix (transposed): M[tx][f][k][n], 16x16.
__global__ __launch_bounds__(256) void build_ifilt_rowG(
    const float* __restrict__ g0a, const float* __restrict__ g0b,
    const float* __restrict__ g1a, const float* __restrict__ g1b,
    float* __restrict__ M, int W) {
  const int tx = blockIdx.x, f = blockIdx.y;
  float ha[10], hb[10];
#pragma unroll
  for (int m = 0; m < 10; ++m) {
    ha[m] = f ? g1b[m] : g0b[m];
    hb[m] = f ? g1a[m] : g0a[m];
  }
  const int p0 = tx << 4;
  const int w0 = iclamp((p0 >> 1) - 4, 0, W - 16);
  const int idx = threadIdx.x;
  const int k = idx >> 4, n = idx & 15;
  M[(((size_t)tx * 2 + f) * 16 + k) * 16 + n] =
      ifilt_coef(p0 + n, w0 + k, W, ha, hb, f != 0);
}

// Level-1 column matrix: M[ty][f][row][k], 16x24 (pitch 24).
__global__ __launch_bounds__(256) void build_j1_colF(
    const float* __restrict__ g0o, const float* __restrict__ g1o,
    float* __restrict__ M, int r) {
  const int ty = blockIdx.x, f = blockIdx.y;
  float h0[7], h1[5];
#pragma unroll
  for (int m = 0; m < 7; ++m) h0[m] = g0o[m];
#pragma unroll
  for (int m = 0; m < 5; ++m) h1[m] = g1o[m];
  const int o0 = ty << 4;
  const int w0 = iclamp(o0 - 4, 0, r - 24);
  for (int idx = threadIdx.x; idx < 16 * 24; idx += 256) {
    const int row = idx / 24, k = idx % 24;
    const float v = f ? filt_coef<5>(o0 + row, w0 + k, r, h1)
                      : filt_coef<7>(o0 + row, w0 + k, r, h0);
    M[((size_t)ty * 2 + f) * 384 + row * 24 + k] = v;
  }
}

// Level-1 row matrix (transposed): M[tx][f][k][n], 24x16 (pitch 16).
__global__ __launch_bounds__(256) void build_j1_rowG(
    const float* __restrict__ g0o, const float* __restrict__ g1o,
    float* __restrict__ M, int W) {
  const int tx = blockIdx.x, f = blockIdx.y;
  float h0[7], h1[5];
#pragma unroll
  for (int m = 0; m < 7; ++m) h0[m] = g0o[m];
#pragma unroll
  for (int m = 0; m < 5; ++m) h1[m] = g1o[m];
  const int p0 = tx << 4;
  const int w0 = iclamp(p0 - 4, 0, W - 24);
  for (int idx = threadIdx.x; idx < 24 * 16; idx += 256) {
    const int k = idx >> 4, n = idx & 15;
    const float v = f ? filt_coef<5>(p0 + n, w0 + k, W, h1)
                      : filt_coef<7>(p0 + n, w0 + k, W, h0);
    M[((size_t)tx * 2 + f) * 384 + k * 16 + n] = v;
  }
}

// ===========================================================================
// Hot tile kernels: one wave = one 16x16 output tile. Dimensions are
// template constants -> all strides fold into immediates / IOFFSETs.
// ===========================================================================

// Column pass, q-shift levels:  y1 = colifilt(ll,lo) + colifilt(lh,hi)
//                               y2 = colifilt(hl,lo) + colifilt(hh,hi)
template <int R>
__global__ __launch_bounds__(32) void col_ifilt_kernel(
    const float* __restrict__ ll, const float* __restrict__ yh,
    const float* __restrict__ Fmat, float* __restrict__ y1,
    float* __restrict__ y2) {
  constexpr int TILES_X = R / 16;
  constexpr int H2 = R / 2;
  constexpr int SB = H2 * H2 * 2;  // subband size in floats
  const int lane = threadIdx.x;
  const int tx = blockIdx.x % TILES_X, ty = blockIdx.x / TILES_X;
  const int o0 = ty << 4, c0 = tx << 4;
  const int mc = blockIdx.y;
  const int which = blockIdx.z;  // 0 -> y1, 1 -> y2
  const int w0 = iclamp((o0 >> 1) - 4, 0, R - 16);  // always even

  const int hg = lane >> 4, l16 = lane & 15;
  const int bcol = c0 + l16;

  const float* llp = ll + (size_t)mc * (R * R);
  const float* yhp = yh + (size_t)mc * (6 * SB);
  const float* pH1 = yhp + (which ? 1 : 0) * SB;  // hh : lh (w1)
  const float* pH2 = yhp + (which ? 4 : 5) * SB;  //         (w2)
  const float* pL1 = yhp + 2 * SB;                // hl pair (which==1)
  const float* pL2 = yhp + 3 * SB;

  // c2q per-lane constants: even rows use col offset bcol with (+,+),
  // odd rows use bcol^1 with (sgn1, sgn2).
  const float sgn1 = (bcol & 1) ? -SC_HALF_SQRT2 : SC_HALF_SQRT2;
  const float sgn2 = (bcol & 1) ? SC_HALF_SQRT2 : -SC_HALF_SQRT2;
  const unsigned colE = (unsigned)bcol;
  const unsigned colO = (unsigned)(bcol ^ 1);

  // A fragments (banded filter matrices), one v2f load each.
  const float* FL = Fmat + ((size_t)ty * 2 + 0) * 256;
  const float* FH = Fmat + ((size_t)ty * 2 + 1) * 256;
  v2f aL[4], aH[4];
#pragma unroll
  for (int t = 0; t < 4; ++t) {
    const int kb = 4 * t + (hg << 1);
    aL[t] = *reinterpret_cast<const v2f*>(FL + l16 * 16 + kb);
    aH[t] = *reinterpret_cast<const v2f*>(FH + l16 * 16 + kb);
  }

  v8f acc = {0.f, 0.f, 0.f, 0.f, 0.f, 0.f, 0.f, 0.f};
#pragma unroll
  for (int t = 0; t < 4; ++t) {
    const int kb = 4 * t + (hg << 1);
    // y = w0+kb is even; even/odd rows share yi, so offset base = y*H2.
    const unsigned rowb = (unsigned)(w0 + kb) * H2;
    v2f bL, bH;
    if (which == 0) {
      const unsigned l0 = (unsigned)(w0 + kb) * R + colE;
      bL.x = llp[l0];
      bL.y = llp[l0 + R];
    } else {
      bL.x = SC_HALF_SQRT2 * (pL1[rowb + colE] + pL2[rowb + colE]);
      bL.y = sgn1 * pL1[rowb + colO] + sgn2 * pL2[rowb + colO];
    }
    bH.x = SC_HALF_SQRT2 * (pH1[rowb + colE] + pH2[rowb + colE]);
    bH.y = sgn1 * pH1[rowb + colO] + sgn2 * pH2[rowb + colO];
    acc = wmma4(aL[t], bL, acc);
    acc = wmma4(aH[t], bH, acc);
  }

  float* outp = (which ? y2 : y1) + (size_t)mc * (2 * R * R) +
                (size_t)(o0 + 8 * hg) * R + bcol;
#pragma unroll
  for (int v = 0; v < 8; ++v) outp[(unsigned)(v * R)] = acc[v];
}

// Row pass, q-shift levels: out = rowifilt(y1,lo) + rowifilt(y2,hi)
template <int H, int W>
__global__ __launch_bounds__(32) void row_ifilt_kernel(
    const float* __restrict__ y1b, const float* __restrict__ y2b,
    const float* __restrict__ Gmat, float* __restrict__ out) {
  constexpr int TILES_X = (2 * W) / 16;
  const int lane = threadIdx.x;
  const int tx = blockIdx.x % TILES_X, ty = blockIdx.x / TILES_X;
  const int yy0 = ty << 4, p0 = tx << 4;
  const int mc = blockIdx.y;
  const int w0 = iclamp((p0 >> 1) - 4, 0, W - 16);

  __shared__ float t1[16 * 17], t2[16 * 17];
  const float* y1p = y1b + (size_t)mc * (H * W);
  const float* y2p = y2b + (size_t)mc * (H * W);
  {  // 2 lanes per row, 8 cols each; constant sub-offsets fold into IOFFSET
    const int rr = lane >> 1, cb = (lane & 1) * 8;
    const unsigned g = (unsigned)(yy0 + rr) * W + (w0 + cb);
#pragma unroll
    for (int j = 0; j < 8; ++j) {
      t1[rr * 17 + cb + j] = y1p[g + j];
      t2[rr * 17 + cb + j] = y2p[g + j];
    }
  }

  // B fragments (coalesced: consecutive lanes -> consecutive n).
  const int hg = lane >> 4, l16 = lane & 15;
  const float* GL = Gmat + ((size_t)tx * 2 + 0) * 256;
  const float* GH = Gmat + ((size_t)tx * 2 + 1) * 256;
  v2f b1[4], b2[4];
#pragma unroll
  for (int t = 0; t < 4; ++t) {
    const int kb = 4 * t + (hg << 1);
    b1[t].x = GL[kb * 16 + l16];
    b1[t].y = GL[(kb + 1) * 16 + l16];
    b2[t].x = GH[kb * 16 + l16];
    b2[t].y = GH[(kb + 1) * 16 + l16];
  }
  __syncthreads();

  v8f acc = {0.f, 0.f, 0.f, 0.f, 0.f, 0.f, 0.f, 0.f};
#pragma unroll
  for (int t = 0; t < 4; ++t) {
    const int kb = 4 * t + (hg << 1);
    v2f a1, a2;
    a1.x = t1[l16 * 17 + kb];     a1.y = t1[l16 * 17 + kb + 1];
    a2.x = t2[l16 * 17 + kb];     a2.y = t2[l16 * 17 + kb + 1];
    acc = wmma4(a1, b1[t], acc);
    acc = wmma4(a2, b2[t], acc);
  }

  float* outp = out + (size_t)mc * (H * 2 * W) +
                (size_t)(yy0 + 8 * hg) * (2 * W) + (p0 + l16);
#pragma unroll
  for (int v = 0; v < 8; ++v) outp[(unsigned)(v * 2 * W)] = acc[v];
}

// Level-1 column pass: y1 = colfilter(ll,g0o) + colfilter(lh,g1o)
//                      y2 = colfilter(hl,g0o) + colfilter(hh,g1o)
template <int R>
__global__ __launch_bounds__(32) void col_j1_kernel(
    const float* __restrict__ ll, const float* __restrict__ yh,
    const float* __restrict__ Fmat, float* __restrict__ y1,
    float* __restrict__ y2) {
  constexpr int TILES_X = R / 16;
  constexpr int H2 = R / 2;
  constexpr int SB = H2 * H2 * 2;
  const int lane = threadIdx.x;
  const int tx = blockIdx.x % TILES_X, ty = blockIdx.x / TILES_X;
  const int o0 = ty << 4, c0 = tx << 4;
  const int mc = blockIdx.y;
  const int which = blockIdx.z;
  const int w0 = iclamp(o0 - 4, 0, R - 24);  // always even

  const int hg = lane >> 4, l16 = lane & 15;
  const int bcol = c0 + l16;

  const float* llp = ll + (size_t)mc * (R * R);
  const float* yhp = yh + (size_t)mc * (6 * SB);
  const float* pH1 = yhp + (which ? 1 : 0) * SB;
  const float* pH2 = yhp + (which ? 4 : 5) * SB;
  const float* pL1 = yhp + 2 * SB;
  const float* pL2 = yhp + 3 * SB;

  const float sgn1 = (bcol & 1) ? -SC_HALF_SQRT2 : SC_HALF_SQRT2;
  const float sgn2 = (bcol & 1) ? SC_HALF_SQRT2 : -SC_HALF_SQRT2;
  const unsigned colE = (unsigned)bcol;
  const unsigned colO = (unsigned)(bcol ^ 1);

  const float* FL = Fmat + ((size_t)ty * 2 + 0) * 384;  // 16x24, pitch 24
  const float* FH = Fmat + ((size_t)ty * 2 + 1) * 384;
  v2f aL[6], aH[6];
#pragma unroll
  for (int t = 0; t < 6; ++t) {
    const int kb = 4 * t + (hg << 1);
    aL[t] = *reinterpret_cast<const v2f*>(FL + l16 * 24 + kb);
    aH[t] = *reinterpret_cast<const v2f*>(FH + l16 * 24 + kb);
  }

  v8f acc = {0.f, 0.f, 0.f, 0.f, 0.f, 0.f, 0.f, 0.f};
#pragma unroll
  for (int t = 0; t < 6; ++t) {
    const int kb = 4 * t + (hg << 1);
    const unsigned rowb = (unsigned)(w0 + kb) * H2;
    v2f bL, bH;
    if (which == 0) {
      const unsigned l0 = (unsigned)(w0 + kb) * R + colE;
      bL.x = llp[l0];
      bL.y = llp[l0 + R];
    } else {
      bL.x = SC_HALF_SQRT2 * (pL1[rowb + colE] + pL2[rowb + colE]);
      bL.y = sgn1 * pL1[rowb + colO] + sgn2 * pL2[rowb + colO];
    }
    bH.x = SC_HALF_SQRT2 * (pH1[rowb + colE] + pH2[rowb + colE]);
    bH.y = sgn1 * pH1[rowb + colO] + sgn2 * pH2[rowb + colO];
    acc = wmma4(aL[t], bL, acc);
    acc = wmma4(aH[t], bH, acc);
  }

  float* outp = (which ? y2 : y1) + (size_t)mc * (R * R) +
                (size_t)(o0 + 8 * hg) * R + bcol;
#pragma unroll
  for (int v = 0; v < 8; ++v) outp[(unsigned)(v * R)] = acc[v];
}

// Level-1 row pass: out = rowfilter(y1,g0o) + rowfilter(y2,g1o); K = 24.
template <int H, int W>
__global__ __launch_bounds__(32) void row_j1_kernel(
    const float* __restrict__ y1b, const float* __restrict__ y2b,
    const float* __restrict__ Gmat, float* __restrict__ out) {
  constexpr int TILES_X = W / 16;
  const int lane = threadIdx.x;
  const int tx = blockIdx.x % TILES_X, ty = blockIdx.x / TILES_X;
  const int yy0 = ty << 4, p0 = tx << 4;
  const int mc = blockIdx.y;
  const int w0 = iclamp(p0 - 4, 0, W - 24);

  __shared__ float t1[16 * 25], t2[16 * 25];
  const float* y1p = y1b + (size_t)mc * (H * W);
  const float* y2p = y2b + (size_t)mc * (H * W);
  {  // 2 lanes per row, 12 cols each
    const int rr = lane >> 1, cb = (lane & 1) * 12;
    const unsigned g = (unsigned)(yy0 + rr) * W + (w0 + cb);
#pragma unroll
    for (int j = 0; j < 12; ++j) {
      t1[rr * 25 + cb + j] = y1p[g + j];
      t2[rr * 25 + cb + j] = y2p[g + j];
    }
  }

  const int hg = lane >> 4, l16 = lane & 15;
  const float* GL = Gmat + ((size_t)tx * 2 + 0) * 384;  // 24x16, pitch 16
  const float* GH = Gmat + ((size_t)tx * 2 + 1) * 384;
  v2f b1[6], b2[6];
#pragma unroll
  for (int t = 0; t < 6; ++t) {
    const int kb = 4 * t + (hg << 1);
    b1[t].x = GL[kb * 16 + l16];
    b1[t].y = GL[(kb + 1) * 16 + l16];
    b2[t].x = GH[kb * 16 + l16];
    b2[t].y = GH[(kb + 1) * 16 + l16];
  }
  __syncthreads();

  v8f acc = {0.f, 0.f, 0.f, 0.f, 0.f, 0.f, 0.f, 0.f};
#pragma unroll
  for (int t = 0; t < 6; ++t) {
    const int kb = 4 * t + (hg << 1);
    v2f a1, a2;
    a1.x = t1[l16 * 25 + kb];     a1.y = t1[l16 * 25 + kb + 1];
    a2.x = t2[l16 * 25 + kb];     a2.y = t2[l16 * 25 + kb + 1];
    acc = wmma4(a1, b1[t], acc);
    acc = wmma4(a2, b2[t], acc);
  }

  float* outp = out + (size_t)mc * (H * W) +
                (size_t)(yy0 + 8 * hg) * W + (p0 + l16);
#pragma unroll
  for (int v = 0; v < 8; ++v) outp[(unsigned)(v * W)] = acc[v];
}

// ---------------------------------------------------------------------------
// Host launcher.
//   ws: [bufA 64MB][bufB 64MB][bufC 64MB][filter matrices 192KB]
// ---------------------------------------------------------------------------
extern "C" void kernel_launch(void* const* d_in, const int* in_sizes, int n_in,
                              void* d_out, int out_size, void* d_ws,
                              size_t ws_size, hipStream_t stream) {
  (void)in_sizes; (void)n_in; (void)out_size; (void)ws_size;
  const float* yl  = (const float*)d_in[0];   // (4,64,64,64)
  const float* yh1 = (const float*)d_in[1];   // (4,64,6,128,128,2)
  const float* yh2 = (const float*)d_in[2];   // (4,64,6,64,64,2)
  const float* yh3 = (const float*)d_in[3];   // (4,64,6,32,32,2)
  const float* g0o = (const float*)d_in[4];   // 7 taps
  const float* g1o = (const float*)d_in[5];   // 5 taps
  const float* g0a = (const float*)d_in[6];   // 10 taps
  const float* g0b = (const float*)d_in[7];
  const float* g1a = (const float*)d_in[8];
  const float* g1b = (const float*)d_in[9];

  const size_t SEG = (size_t)16 * 1024 * 1024;  // 16M floats = 64 MB
  float* bufA = (float*)d_ws;
  float* bufB = bufA + SEG;
  float* bufC = bufB + SEG;
  float* F3c = bufC + SEG;           //  8*2*256
  float* G3r = F3c + 8 * 2 * 256;    //  8*2*256
  float* F2c = G3r + 8 * 2 * 256;    // 16*2*256
  float* G2r = F2c + 16 * 2 * 256;   // 16*2*256
  float* F1c = G2r + 16 * 2 * 256;   // 16*2*384
  float* G1r = F1c + 16 * 2 * 384;   // 16*2*384

  // --- build all banded filter matrices (tiny; shared by maps/tiles) ---
  build_ifilt_colF<<<dim3(8, 2),  256, 0, stream>>>(g0a, g0b, g1a, g1b, F3c, 64);
  build_ifilt_rowG<<<dim3(8, 2),  256, 0, stream>>>(g0a, g0b, g1a, g1b, G3r, 64);
  build_ifilt_colF<<<dim3(16, 2), 256, 0, stream>>>(g0a, g0b, g1a, g1b, F2c, 128);
  build_ifilt_rowG<<<dim3(16, 2), 256, 0, stream>>>(g0a, g0b, g1a, g1b, G2r, 128);
  build_j1_colF<<<dim3(16, 2),    256, 0, stream>>>(g0o, g1o, F1c, 256);
  build_j1_rowG<<<dim3(16, 2),    256, 0, stream>>>(g0o, g1o, G1r, 256);

  // Level 3: ll 64x64 + yh3 -> 128x128
  col_ifilt_kernel<64><<<dim3(8 * 4, 256, 2), 32, 0, stream>>>(
      yl, yh3, F3c, bufA, bufB);
  row_ifilt_kernel<128, 64><<<dim3(8 * 8, 256, 1), 32, 0, stream>>>(
      bufA, bufB, G3r, bufC);

  // Level 2: ll 128x128 + yh2 -> 256x256
  col_ifilt_kernel<128><<<dim3(16 * 8, 256, 2), 32, 0, stream>>>(
      bufC, yh2, F2c, bufA, bufB);
  row_ifilt_kernel<256, 128><<<dim3(16 * 16, 256, 1), 32, 0, stream>>>(
      bufA, bufB, G2r, bufC);

  // Level 1: ll 256x256 + yh1 -> 256x256 (final)
  col_j1_kernel<256><<<dim3(16 * 16, 256, 2), 32, 0, stream>>>(
      bufC, yh1, F1c, bufA, bufB);
  row_j1_kernel<256, 256><<<dim3(16 * 16, 256, 1), 32, 0, stream>>>(
      bufA, bufB, G1r, (float*)d_out);
}